// Environment_model_8443905704155
// MI455X (gfx1250) — compile-verified
//
#include <hip/hip_runtime.h>

typedef __attribute__((ext_vector_type(2))) float v2f;
typedef __attribute__((ext_vector_type(4))) float v4f;
typedef __attribute__((ext_vector_type(8))) float v8f;

#define B_SZ 8192
#define T_SZ 512
#define H_SZ 7
#define IN_F 5
#define G3   21      // 3*H
#define XPS  32      // padded xp row stride in floats (128 B = 2 cachelines, 16B-aligned)

// ---------------------------------------------------------------------------
// Kernel 1: xp[t_local][b][g] = x[b][t][:] @ W_ih^T + b_ih (+ b_hh for r,z)
// One wave computes a 16(time) x 21(gate) tile via V_WMMA_F32_16X16X4_F32.
// K = 5 features padded to 8 -> 2 K-chunks; N = 21 gates -> 2 N-tiles.
// Fully branchless: all loads unconditional with clamped (in-bounds) indices,
// zeros inserted by value-selects; stores unconditional into 32-col padded
// rows (cols 21..31 are dead padding, never read).
// ---------------------------------------------------------------------------
__global__ void __launch_bounds__(128)
gru_xproj_wmma(const float* __restrict__ x,
               const float* __restrict__ W_ih,
               const float* __restrict__ b_ih,
               const float* __restrict__ b_hh,
               float* __restrict__ xp,
               int t_base, int tpb_shift)
{
  const int lane = threadIdx.x & 31;
  const int wave = (blockIdx.x << 2) | (threadIdx.x >> 5);
  const int b    = wave >> tpb_shift;                       // batch row
  const int t0l  = (wave & ((1 << tpb_shift) - 1)) << 4;    // chunk-local t of tile

  const int  m  = lane & 15;                // row (time) / col (gate) index
  const bool lo = (lane < 16);
  const int  kb = lo ? 0 : 2;               // K-pair this lane supplies

  // ---- A operand: 16 time-rows x 4 features, two K-chunks (feat 4..7 zero-pad)
  const long xrow = ((long)b * T_SZ + (t_base + t0l + m)) * IN_F;
  const float xa  = x[xrow + kb];
  const float xb2 = x[xrow + kb + 1];
  const float x4  = x[xrow + 4];            // unconditional, always in-bounds
  v2f a0, a1;
  a0.x = xa;  a0.y = xb2;
  a1.x = lo ? x4 : 0.0f;                    // chunk1 supplies only K=4
  a1.y = 0.0f;

  // ---- B operand: W_ih^T (K x gate); tile0 = gates 0..15, tile1 = gates 16..20
  const int  g0  = m;
  const int  g1  = 16 + m;
  const bool gv  = (g1 < G3);
  const int  g1c = gv ? g1 : (G3 - 1);      // clamped: loads always in-bounds
  const float w00 = W_ih[g0 * IN_F + kb];
  const float w01 = W_ih[g0 * IN_F + kb + 1];
  const float w04 = W_ih[g0 * IN_F + 4];
  const float w10 = W_ih[g1c * IN_F + kb];
  const float w11 = W_ih[g1c * IN_F + kb + 1];
  const float w14 = W_ih[g1c * IN_F + 4];
  v2f b00, b10, b01, b11;
  b00.x = w00;                      b00.y = w01;
  b10.x = lo ? w04 : 0.0f;          b10.y = 0.0f;
  b01.x = gv ? w10 : 0.0f;          b01.y = gv ? w11 : 0.0f;
  b11.x = (gv && lo) ? w14 : 0.0f;  b11.y = 0.0f;

  // ---- C init: fuse b_ih (all gates) and b_hh (r,z gates only; n-gate b_hh
  //      must stay inside r*hn, handled in the recurrence kernel).
  //      Loads hoisted unconditionally (in-bounds); selects act on values.
  const float bi0 = b_ih[g0];
  const float bh0 = b_hh[g0];               // g0 <= 15 < 21: in-bounds
  const float bi1 = b_ih[g1c];              // clamped: in-bounds
  const float bias0 = bi0 + ((g0 < 2 * H_SZ) ? bh0 : 0.0f);
  const float bias1 = gv ? bi1 : 0.0f;      // tile1 gates are all n-gates: no b_hh
  v8f c0, c1;
  #pragma unroll
  for (int i = 0; i < 8; ++i) { c0[i] = bias0; c1[i] = bias1; }

  c0 = __builtin_amdgcn_wmma_f32_16x16x4_f32(false, a0, false, b00, (short)0, c0, false, false);
  c0 = __builtin_amdgcn_wmma_f32_16x16x4_f32(false, a1, false, b10, (short)0, c0, false, false);
  c1 = __builtin_amdgcn_wmma_f32_16x16x4_f32(false, a0, false, b01, (short)0, c1, false, false);
  c1 = __builtin_amdgcn_wmma_f32_16x16x4_f32(false, a1, false, b11, (short)0, c1, false, false);

  // ---- store D time-major: xp[(t_local)*B + b]*XPS + gate, unconditional.
  //      D layout: VGPR v holds M=v (lanes 0-15) / M=v+8 (lanes 16-31), N=lane%16
  const int mrow = lo ? 0 : 8;
  #pragma unroll
  for (int v = 0; v < 8; ++v) {
    const long base = ((long)(t0l + v + mrow) * B_SZ + b) * XPS;
    xp[base + m]      = c0[v];   // gates 0..15 : 64B contiguous per half-wave
    xp[base + 16 + m] = c1[v];   // gates 16..20 (+ dead padding cols 21..31)
  }
}

// ---------------------------------------------------------------------------
// Kernel 2: sequential GRU over a chunk of timesteps; one lane per batch
// element (h, W_hh in registers), output heads fused (weights in LDS),
// reading xp coalesced (time-major, 128 B rows -> 5x b128 + 1x b32).
// ---------------------------------------------------------------------------
__global__ void __launch_bounds__(256)
gru_recur_fused(const float* __restrict__ xp,
                const float* __restrict__ W_hh,
                const float* __restrict__ b_hh,
                const float* __restrict__ W_h0,
                const float* __restrict__ W_m,
                const float* __restrict__ b_m,
                const float* __restrict__ W_r,
                const float* __restrict__ b_r,
                float* __restrict__ h_buf,
                float* __restrict__ out,
                int t_base, int t_len, int first, int last)
{
  __shared__ float sW[32]; // [0..20]=W_m, [21..23]=b_m, [24..30]=W_r, [31]=b_r
  const int tid = threadIdx.x;
  if      (tid < 21) sW[tid] = W_m[tid];
  else if (tid < 24) sW[tid] = b_m[tid - 21];
  else if (tid < 31) sW[tid] = W_r[tid - 24];
  else if (tid == 31) sW[31] = b_r[0];
  __syncthreads();

  const int b = blockIdx.x * 256 + tid;

  float wh[G3 * H_SZ];                      // W_hh, uniform -> scalar-loaded
  #pragma unroll
  for (int i = 0; i < G3 * H_SZ; ++i) wh[i] = W_hh[i];
  float bn[H_SZ];                           // b_hh for n-gates
  #pragma unroll
  for (int j = 0; j < H_SZ; ++j) bn[j] = b_hh[2 * H_SZ + j];

  float h[H_SZ];
  if (first) {
    #pragma unroll
    for (int j = 0; j < H_SZ; ++j) h[j] = W_h0[j];
  } else {
    #pragma unroll
    for (int j = 0; j < H_SZ; ++j) h[j] = h_buf[b * 8 + j];
  }

  const float* xrow = xp + (long)b * XPS;
  for (int tl = 0; tl < t_len; ++tl) {
    const float* xr = xrow + (long)tl * B_SZ * XPS;
    if (tl + 1 < t_len)
      __builtin_prefetch(xr + (long)B_SZ * XPS, 0, 0);   // next step's xp row

    // 21 gate pre-activations: 5x 16B vector loads + 1 scalar (row 16B-aligned)
    const v4f q0 = *(const v4f*)(xr + 0);
    const v4f q1 = *(const v4f*)(xr + 4);
    const v4f q2 = *(const v4f*)(xr + 8);
    const v4f q3 = *(const v4f*)(xr + 12);
    const v4f q4 = *(const v4f*)(xr + 16);
    const float x20 = xr[20];

    float aR[H_SZ] = {q0.x, q0.y, q0.z, q0.w, q1.x, q1.y, q1.z}; // + b_ih + b_hh
    float aZ[H_SZ] = {q1.w, q2.x, q2.y, q2.z, q2.w, q3.x, q3.y}; // + b_ih + b_hh
    float xn[H_SZ] = {q3.z, q3.w, q4.x, q4.y, q4.z, q4.w, x20};  // + b_ih
    float aN[H_SZ];
    #pragma unroll
    for (int j = 0; j < H_SZ; ++j) aN[j] = bn[j];

    #pragma unroll
    for (int j = 0; j < H_SZ; ++j) {
      #pragma unroll
      for (int k = 0; k < H_SZ; ++k) {
        aR[j] = fmaf(wh[j * H_SZ + k],              h[k], aR[j]);
        aZ[j] = fmaf(wh[(H_SZ + j) * H_SZ + k],     h[k], aZ[j]);
        aN[j] = fmaf(wh[(2 * H_SZ + j) * H_SZ + k], h[k], aN[j]);
      }
    }
    #pragma unroll
    for (int j = 0; j < H_SZ; ++j) {
      const float r = 1.0f / (1.0f + __expf(-aR[j]));
      const float z = 1.0f / (1.0f + __expf(-aZ[j]));
      const float n = tanhf(xn[j] + r * aN[j]);
      h[j] = n + z * (h[j] - n);   // (1-z)*n + z*h
    }

    // fused output heads
    float o0 = sW[21], o1 = sW[22], o2 = sW[23], orr = sW[31];
    #pragma unroll
    for (int k = 0; k < H_SZ; ++k) {
      o0  = fmaf(sW[k],            h[k], o0);
      o1  = fmaf(sW[H_SZ + k],     h[k], o1);
      o2  = fmaf(sW[2 * H_SZ + k], h[k], o2);
      orr = fmaf(sW[24 + k],       h[k], orr);
    }
    const int  tg = t_base + tl;
    const long ob = ((long)b * T_SZ + tg) * 3;
    out[ob]     = o0;
    out[ob + 1] = o1;
    out[ob + 2] = o2;
    out[(long)B_SZ * T_SZ * 3 + (long)b * T_SZ + tg] = orr;
  }

  #pragma unroll
  for (int j = 0; j < H_SZ; ++j) h_buf[b * 8 + j] = h[j];
  if (last) {
    #pragma unroll
    for (int j = 0; j < H_SZ; ++j)
      out[(long)B_SZ * T_SZ * 4 + b * H_SZ + j] = h[j];   // h_last [1,B,H]
  }
}

// ---------------------------------------------------------------------------
extern "C" void kernel_launch(void* const* d_in, const int* in_sizes, int n_in,
                              void* d_out, int out_size, void* d_ws, size_t ws_size,
                              hipStream_t stream)
{
  (void)in_sizes; (void)n_in; (void)out_size;
  const float* x    = (const float*)d_in[0];
  const float* W_ih = (const float*)d_in[2];
  const float* W_hh = (const float*)d_in[3];
  const float* b_ih = (const float*)d_in[4];
  const float* b_hh = (const float*)d_in[5];
  const float* W_h0 = (const float*)d_in[6];
  const float* W_m  = (const float*)d_in[7];
  const float* b_m  = (const float*)d_in[8];
  const float* W_r  = (const float*)d_in[9];
  const float* b_r  = (const float*)d_in[10];
  float* out = (float*)d_out;

  float* h_buf = (float*)d_ws;                                   // B*8 floats
  float* xpbuf = (float*)((char*)d_ws + (size_t)B_SZ * 8 * 4);   // +256 KB

  // Chunk of timesteps whose xp buffer fits in ws. Cap at 128 so the
  // producer->consumer xp traffic (128*8192*128B = 134 MB) stays resident
  // in the 192 MB L2 and never round-trips HBM.
  size_t avail = (ws_size > (size_t)B_SZ * 8 * 4) ? ws_size - (size_t)B_SZ * 8 * 4 : 0;
  int tc = 16;
  const int cand[3] = {128, 64, 32};
  for (int i = 0; i < 3; ++i) {
    if ((size_t)cand[i] * B_SZ * XPS * 4 <= avail) { tc = cand[i]; break; }
  }
  int sh = 0;
  while ((1 << sh) < (tc >> 4)) ++sh;   // tiles-per-batch-row shift

  for (int t0 = 0; t0 < T_SZ; t0 += tc) {
    const int nwaves = B_SZ * (tc >> 4);          // one 16x21 tile per wave
    gru_xproj_wmma<<<nwaves >> 2, 128, 0, stream>>>(x, W_ih, b_ih, b_hh,
                                                    xpbuf, t0, sh);
    gru_recur_fused<<<B_SZ / 256, 256, 0, stream>>>(xpbuf, W_hh, b_hh, W_h0,
                                                    W_m, b_m, W_r, b_r,
                                                    h_buf, out, t0, tc,
                                                    t0 == 0, (t0 + tc) == T_SZ);
  }
}